// MetaDataTokens_47201690583127
// MI455X (gfx1250) — compile-verified
//
#include <hip/hip_runtime.h>
#include <hip/hip_bf16.h>
#include <math.h>

// Problem constants (match reference)
#define SB 16384   // batch
#define SS 20      // seq len
#define SD 128     // embed dim
#define SV 12      // vocab
#define FEPS 1e-5f

typedef __attribute__((ext_vector_type(2))) float v2f;
typedef __attribute__((ext_vector_type(8))) float v8f;

// LDS strides
#define XN_STRIDE 136   // 4-pad | 128 | 4-pad  -> conv halo, no bounds checks
#define GL_STRIDE 132

__global__ __launch_bounds__(128)
void fused_metatokens_kernel(
    const int*   __restrict__ tokens,   // [B,S]
    const float* __restrict__ emb,      // [V,D]
    const float* __restrict__ pre_w,    // [S]
    const float* __restrict__ w0, const float* __restrict__ b0,   // [S,1,3],[S]
    const float* __restrict__ w1, const float* __restrict__ b1,
    const float* __restrict__ w2, const float* __restrict__ b2,
    const float* __restrict__ gate_w,   // [S,S]  g[t]=sum_s xn[s]*gate_w[t,s]
    const float* __restrict__ gate_b,   // [S]
    const float* __restrict__ post_w,   // [S]
    const float* __restrict__ logit_w,  // [1,S,1]
    float*       __restrict__ out)      // [B,1,D]
{
    __shared__ int   tok[SS];
    __shared__ float gb[SS];
    __shared__ float Gp[32 * SS];            // gate_w padded to 32 rows (t), 20 cols (s)
    __shared__ float xn[SS * XN_STRIDE];     // x_norm tile with +-4 halo
    __shared__ float gl[SS * GL_STRIDE];     // silu(gate) tile

    const int tid = threadIdx.x;             // == d column, 0..127
    const int bidx = blockIdx.x;             // == batch element

    // ---- setup: stage tokens / gate bias / padded gate matrix / halo zeros ----
    if (tid < SS) {
        tok[tid] = tokens[bidx * SS + tid];
        gb[tid]  = gate_b[tid];
    }
    for (int e = tid; e < 32 * SS; e += 128) {
        int t = e / SS, s = e - t * SS;
        Gp[e] = (t < SS) ? gate_w[t * SS + s] : 0.0f;
    }
    for (int e = tid; e < SS * 8; e += 128) {        // zero conv halos
        int s = e >> 3, p = e & 7;
        xn[s * XN_STRIDE + ((p < 4) ? p : (128 + p))] = 0.0f;
    }
    __syncthreads();

    // ---- embedding gather + pre-RMSNorm over S (thread-local per (b,d)) ----
    float x[SS];
    float ss = 0.0f;
    #pragma unroll
    for (int s = 0; s < SS; ++s) {
        x[s] = emb[tok[s] * SD + tid];
        ss += x[s] * x[s];
    }
    const float inv = rsqrtf(ss * (1.0f / SS) + FEPS);
    #pragma unroll
    for (int s = 0; s < SS; ++s) {
        xn[s * XN_STRIDE + 4 + tid] = x[s] * inv * pre_w[s];
    }
    __syncthreads();

    // ---- gate GEMM via V_WMMA_F32_16X16X4_F32 ----
    // per batch: gate_w[20x20] x xn[20x128]; M=20 -> 2 tiles (padded), K=20 -> 5 steps,
    // N=128 -> 8 tiles, 2 per wave (4 waves/block).
    const int lane = tid & 31;
    const int wv   = tid >> 5;      // wave id 0..3
    const int lh   = lane >> 4;     // lane half
    const int l15  = lane & 15;

    #pragma unroll
    for (int ni = 0; ni < 2; ++ni) {
        const int nt   = 2 * wv + ni;
        const int dcol = nt * 16 + l15;
        #pragma unroll
        for (int mt = 0; mt < 2; ++mt) {
            v8f acc = {};
            #pragma unroll
            for (int kk = 0; kk < 5; ++kk) {
                // A 16x4 (MxK): lanes 0-15 hold K=0,1 in v0,v1; lanes 16-31 hold K=2,3
                const int arow = mt * 16 + l15;
                const int kb   = kk * 4 + lh * 2;
                v2f a, bf;
                a.x  = Gp[arow * SS + kb];
                a.y  = Gp[arow * SS + kb + 1];
                // B 4x16 (KxN): N on lanes (lane&15), K split lane-half/VGPR
                bf.x = xn[(kb + 0) * XN_STRIDE + 4 + dcol];
                bf.y = xn[(kb + 1) * XN_STRIDE + 4 + dcol];
                acc = __builtin_amdgcn_wmma_f32_16x16x4_f32(
                          false, a, false, bf, (short)0, acc, false, false);
            }
            // D 16x16: lane<16 -> row = 16*mt + r ; lane>=16 -> row = 16*mt + r + 8
            #pragma unroll
            for (int r = 0; r < 8; ++r) {
                const int t = mt * 16 + r + lh * 8;
                if (t < SS) {
                    // fused bias + SiLU; native v_rcp_f32 (TRANS, co-executes with
                    // the WMMA/XDL stream) instead of the precise-div sequence.
                    float v  = acc[r] + gb[t];
                    float sg = __builtin_amdgcn_rcpf(1.0f + __expf(-v));
                    gl[t * GL_STRIDE + dcol] = v * sg;
                }
            }
        }
    }
    __syncthreads();

    // ---- multi-dilation depthwise conv + gate + post-RMSNorm + logit head ----
    float z[SS];
    float ss2 = 0.0f;
    #pragma unroll
    for (int s = 0; s < SS; ++s) {
        const float* row = &xn[s * XN_STRIDE + 4 + tid];
        float c = b0[s] + b1[s] + b2[s];
        c += w0[s * 3 + 0] * row[-1] + w0[s * 3 + 1] * row[0] + w0[s * 3 + 2] * row[1];
        c += w1[s * 3 + 0] * row[-2] + w1[s * 3 + 1] * row[0] + w1[s * 3 + 2] * row[2];
        c += w2[s * 3 + 0] * row[-4] + w2[s * 3 + 1] * row[0] + w2[s * 3 + 2] * row[4];
        const float zz = c * gl[s * GL_STRIDE + tid];
        z[s] = zz;
        ss2 += zz * zz;
    }
    const float inv2 = rsqrtf(ss2 * (1.0f / SS) + FEPS);
    float o = 0.0f;
    #pragma unroll
    for (int s = 0; s < SS; ++s) {
        o += (x[s] + z[s] * inv2 * post_w[s]) * logit_w[s];
    }
    out[bidx * SD + tid] = o;
}

extern "C" void kernel_launch(void* const* d_in, const int* in_sizes, int n_in,
                              void* d_out, int out_size, void* d_ws, size_t ws_size,
                              hipStream_t stream) {
    // setup_inputs() order:
    // 0 tokens, 1 number_log(unused), 2 emb, 3 pre_w, 4 w0, 5 b0, 6 w1, 7 b1,
    // 8 w2, 9 b2, 10 gate_w, 11 gate_b, 12 post_w, 13 logit_w
    const int*   tokens = (const int*)  d_in[0];
    const float* emb    = (const float*)d_in[2];
    const float* pre_w  = (const float*)d_in[3];
    const float* w0     = (const float*)d_in[4];
    const float* b0     = (const float*)d_in[5];
    const float* w1     = (const float*)d_in[6];
    const float* b1     = (const float*)d_in[7];
    const float* w2     = (const float*)d_in[8];
    const float* b2     = (const float*)d_in[9];
    const float* gate_w = (const float*)d_in[10];
    const float* gate_b = (const float*)d_in[11];
    const float* post_w = (const float*)d_in[12];
    const float* logit_w= (const float*)d_in[13];
    float* out = (float*)d_out;

    const int batch = in_sizes[0] / SS;   // 16384
    fused_metatokens_kernel<<<dim3(batch), dim3(128), 0, stream>>>(
        tokens, emb, pre_w, w0, b0, w1, b1, w2, b2,
        gate_w, gate_b, post_w, logit_w, out);
}